// SetCriterion_12051678232719
// MI455X (gfx1250) — compile-verified
//
#include <hip/hip_runtime.h>
#include <hip/hip_bf16.h>
#include <stdint.h>

#define BATCH 64
#define NQ    900
#define NT    30
#define NCLS  1001          // NUM_CLASSES + 1
#define QN    (NQ * NT)     // 27000
#define MASKV 1e9f

// ---------------------------------------------------------------------------
// helpers
// ---------------------------------------------------------------------------
__device__ __forceinline__ float clamp01(float x) {
    return fminf(fmaxf(x, 0.0f), 1.0f);
}

__device__ __forceinline__ float giou_xyxy1(const float p[4], const float t[4]) {
    float lx = fmaxf(p[0], t[0]), ly = fmaxf(p[1], t[1]);
    float rx = fminf(p[2], t[2]), ry = fminf(p[3], t[3]);
    float iw = fmaxf(rx - lx, 0.0f), ih = fmaxf(ry - ly, 0.0f);
    float inter = iw * ih;
    float a1 = fmaxf(p[2] - p[0], 0.0f) * fmaxf(p[3] - p[1], 0.0f);
    float a2 = fmaxf(t[2] - t[0], 0.0f) * fmaxf(t[3] - t[1], 0.0f);
    float uni = a1 + a2 - inter;
    float iou = inter / fmaxf(uni, 1e-6f);
    float cx0 = fminf(p[0], t[0]), cy0 = fminf(p[1], t[1]);
    float cx1 = fmaxf(p[2], t[2]), cy1 = fmaxf(p[3], t[3]);
    float cw = fmaxf(cx1 - cx0, 0.0f), ch = fmaxf(cy1 - cy0, 0.0f);
    float ac = cw * ch;
    return iou - (ac - uni) / fmaxf(ac, 1e-6f);
}

// ---------------------------------------------------------------------------
// Kernel 1: fused logsumexp + cost-matrix row. One 256-thread block per row.
// Single HBM pass over the 230MB logits tensor (the dominant cost).
// Loads are issued back-to-back into registers (pipelined), then staged to
// LDS so threads 0..29 can gather x[label] for the class cost.
// ---------------------------------------------------------------------------
__global__ void __launch_bounds__(256)
k_lse_cost(const float* __restrict__ logits,
           const float* __restrict__ pboxes,
           const int*   __restrict__ labels,
           const float* __restrict__ tboxes,
           float* __restrict__ lse,
           float* __restrict__ pxyxy,
           float* __restrict__ Cmat)
{
    __shared__ float xrow[NCLS];
    __shared__ float red[8];
    __shared__ float bval[2];   // [0]=row max, [1]=lse
    __shared__ float px[4];

    const int row = blockIdx.x;              // b*NQ + q
    const int b   = row / NQ;
    const int tid = threadIdx.x;             // 256 threads
    const int lane = tid & 31, wid = tid >> 5;
    const float* xr = logits + (size_t)row * NCLS;

    // issue all 4 loads, then stores (768 = 3*256 < 1001; tail is 233 wide)
    float v0 = xr[tid];
    float v1 = xr[tid + 256];
    float v2 = xr[tid + 512];
    float v3 = (tid < NCLS - 768) ? xr[tid + 768] : -INFINITY;
    xrow[tid]       = v0;
    xrow[tid + 256] = v1;
    xrow[tid + 512] = v2;
    if (tid < NCLS - 768) xrow[tid + 768] = v3;

    float m = fmaxf(fmaxf(v0, v1), fmaxf(v2, v3));
    // block max
    for (int o = 16; o; o >>= 1) m = fmaxf(m, __shfl_down(m, o, 32));
    if (lane == 0) red[wid] = m;
    __syncthreads();
    if (tid == 0) {
        float r = red[0];
        for (int w = 1; w < 8; ++w) r = fmaxf(r, red[w]);
        bval[0] = r;
    }
    __syncthreads();
    const float mx = bval[0];

    // expf(-INF - mx) == 0, so the tail term is safe unconditionally
    float s = expf(v0 - mx) + expf(v1 - mx) + expf(v2 - mx) + expf(v3 - mx);
    for (int o = 16; o; o >>= 1) s += __shfl_down(s, o, 32);
    if (lane == 0) red[wid] = s;
    __syncthreads();
    if (tid == 0) {
        float r = 0.0f;
        for (int w = 0; w < 8; ++w) r += red[w];
        float L = mx + logf(r);
        bval[1] = L;
        lse[row] = L;
    }
    __syncthreads();
    if (tid == 0) {
        float cx = pboxes[(size_t)row * 4 + 0];
        float cy = pboxes[(size_t)row * 4 + 1];
        float w  = pboxes[(size_t)row * 4 + 2];
        float h  = pboxes[(size_t)row * 4 + 3];
        px[0] = clamp01(cx - 0.5f * w);
        px[1] = clamp01(cy - 0.5f * h);
        px[2] = clamp01(cx + 0.5f * w);
        px[3] = clamp01(cy + 0.5f * h);
        pxyxy[(size_t)row * 4 + 0] = px[0];
        pxyxy[(size_t)row * 4 + 1] = px[1];
        pxyxy[(size_t)row * 4 + 2] = px[2];
        pxyxy[(size_t)row * 4 + 3] = px[3];
    }
    __syncthreads();
    const float L = bval[1];
    if (tid < NT) {
        int lab = labels[b * NT + tid];
        float ccls = -expf(xrow[lab] - L);               // -softmax prob
        const float* tb = tboxes + ((size_t)(b * NT + tid)) * 4;
        float p4[4] = { px[0], px[1], px[2], px[3] };
        float t4[4] = { tb[0], tb[1], tb[2], tb[3] };
        float l1 = fabsf(p4[0] - t4[0]) + fabsf(p4[1] - t4[1]) +
                   fabsf(p4[2] - t4[2]) + fabsf(p4[3] - t4[3]);
        float g = giou_xyxy1(p4, t4);
        Cmat[(size_t)row * NT + tid] = 1.0f * ccls + 5.0f * l1 + 2.0f * (-g);
    }
}

// ---------------------------------------------------------------------------
// Kernel 2: greedy matcher. One 1024-thread block per batch.
// Cost tile (27000 f32 = 108KB, fits the 320KB WGP LDS) staged into LDS via
// the CDNA5 Tensor Data Mover (tensor_load_to_lds + s_wait_tensorcnt), then
// 30 serial rounds of block-wide argmin (first-index tie-break, matching
// np.argmin) + row/col masking. Scan uses ds_load_b128 (float4, QN%4==0).
// ---------------------------------------------------------------------------
#if defined(__AMDGCN__)
typedef unsigned int u32x4 __attribute__((ext_vector_type(4)));
typedef int          i32x8 __attribute__((ext_vector_type(8)));
typedef int          i32x4 __attribute__((ext_vector_type(4)));
#endif

__global__ void __launch_bounds__(1024)
k_match(const float* __restrict__ Cmat,
        const int*   __restrict__ labels,
        int* __restrict__ pi, int* __restrict__ ti,
        int* __restrict__ tc)
{
    extern __shared__ float smem[];
    float* Cs = smem;                       // QN floats (16B aligned base)
    float* rv = smem + QN;                  // 32
    int*   ri = (int*)(rv + 32);            // 32
    int*   pq = ri + 32;                    // 30
    int*   pt = pq + NT;                    // 30
    int*   bc = pt + NT;                    // 2

    const int b   = blockIdx.x;
    const int tid = threadIdx.x;
    const int nt  = blockDim.x;             // 1024
    const int lane = tid & 31, wid = tid >> 5;
    const int nw = nt >> 5;                 // 32 waves
    const float* Cb = Cmat + (size_t)b * QN;

#if defined(__gfx1250__) && __has_builtin(__builtin_amdgcn_tensor_load_to_lds)
    if (tid < 32) {
        // D# group0: count=1 (valid), lds_addr, global_addr[56:0], type=2
        unsigned long long ga = (unsigned long long)(const void*)Cb;
        unsigned int ldsoff = (unsigned int)(size_t)(void*)Cs;  // flat LDS addr low 32b = offset
        u32x4 g0 = { 1u,
                     ldsoff,
                     (unsigned int)(ga & 0xFFFFFFFFu),
                     (unsigned int)(((ga >> 32) & 0x01FFFFFFu) | 0x80000000u) };
        // D# group1: wg_mask=0, data_size=2 (4B), tensor_dim0=QN, tensor_dim1=1,
        // tile_dim0=QN, tile_dim1=1, stride0=QN, stride1=QN
        i32x8 g1 = { (int)0x00020000u,
                     (int)(((unsigned)QN & 0xFFFFu) << 16),
                     (int)((((unsigned)QN >> 16) & 0xFFFFu) | (1u << 16)),
                     (int)(((unsigned)QN & 0xFFFFu) << 16),
                     (int)1,
                     (int)QN,
                     (int)(((unsigned)QN & 0xFFFFu) << 16),
                     (int)(((unsigned)QN >> 16) & 0xFFFFu) };
        i32x4 g2 = { 0, 0, 0, 0 };
        i32x4 g3 = { 0, 0, 0, 0 };
        i32x8 g4 = { 0, 0, 0, 0, 0, 0, 0, 0 };   // 6-arg toolchain: extra group (zero)
        __builtin_amdgcn_tensor_load_to_lds(g0, g1, g2, g3, g4, 0);
#if __has_builtin(__builtin_amdgcn_s_wait_tensorcnt)
        __builtin_amdgcn_s_wait_tensorcnt(0);
#else
        asm volatile("s_wait_tensorcnt 0x0" ::: "memory");
#endif
    }
#else
    for (int i = tid; i < QN; i += nt) Cs[i] = Cb[i];
#endif
    __syncthreads();

    const float4* Cv = (const float4*)Cs;
    const int nv4 = QN / 4;                 // 6750 vec4 elements

    for (int r = 0; r < NT; ++r) {
        float best = INFINITY;
        int   bi   = QN;
        for (int j = tid; j < nv4; j += nt) {
            float4 vv = Cv[j];
            int i0 = j * 4;
            if (vv.x < best || (vv.x == best && i0     < bi)) { best = vv.x; bi = i0;     }
            if (vv.y < best || (vv.y == best && i0 + 1 < bi)) { best = vv.y; bi = i0 + 1; }
            if (vv.z < best || (vv.z == best && i0 + 2 < bi)) { best = vv.z; bi = i0 + 2; }
            if (vv.w < best || (vv.w == best && i0 + 3 < bi)) { best = vv.w; bi = i0 + 3; }
        }
        for (int o = 16; o; o >>= 1) {
            float ov = __shfl_down(best, o, 32);
            int   oi = __shfl_down(bi, o, 32);
            if (ov < best || (ov == best && oi < bi)) { best = ov; bi = oi; }
        }
        if (lane == 0) { rv[wid] = best; ri[wid] = bi; }
        __syncthreads();
        if (tid == 0) {
            float wb = rv[0]; int wi = ri[0];
            for (int w = 1; w < nw; ++w) {
                float ov = rv[w]; int oi = ri[w];
                if (ov < wb || (ov == wb && oi < wi)) { wb = ov; wi = oi; }
            }
            int q = wi / NT, t = wi % NT;
            pq[r] = q; pt[r] = t;
            bc[0] = q; bc[1] = t;
        }
        __syncthreads();
        int q = bc[0], t = bc[1];
        if (tid < NT) Cs[q * NT + tid] = MASKV;                 // mask row
        for (int qq = tid; qq < NQ; qq += nt) Cs[qq * NT + t] = MASKV;  // mask col
        __syncthreads();
    }

    // write results + target-class array
    for (int q = tid; q < NQ; q += nt) tc[b * NQ + q] = NCLS - 1;   // no-object
    __syncthreads();
    if (tid < NT) {
        int q = pq[tid], t = pt[tid];
        pi[b * NT + tid] = q;
        ti[b * NT + tid] = t;
        tc[b * NQ + q] = labels[b * NT + t];
    }
}

// ---------------------------------------------------------------------------
// Kernel 3: weighted CE per batch (torch weighted-CE, per-image normalized).
// ---------------------------------------------------------------------------
__global__ void __launch_bounds__(256)
k_ce(const float* __restrict__ logits,
     const float* __restrict__ lse,
     const int*   __restrict__ tc,
     float* __restrict__ ce_b)
{
    __shared__ float r1[8], r2[8];
    const int b = blockIdx.x;
    const int tid = threadIdx.x;        // 256
    const int lane = tid & 31, wid = tid >> 5;
    float swn = 0.0f, sw = 0.0f;
    for (int q = tid; q < NQ; q += 256) {
        int row = b * NQ + q;
        int c = tc[row];
        float x = logits[(size_t)row * NCLS + c];
        float nll = lse[row] - x;
        float w = (c == NCLS - 1) ? 0.1f : 1.0f;
        swn += w * nll;
        sw  += w;
    }
    for (int o = 16; o; o >>= 1) {
        swn += __shfl_down(swn, o, 32);
        sw  += __shfl_down(sw,  o, 32);
    }
    if (lane == 0) { r1[wid] = swn; r2[wid] = sw; }
    __syncthreads();
    if (tid == 0) {
        float a = 0.0f, c = 0.0f;
        for (int w = 0; w < 8; ++w) { a += r1[w]; c += r2[w]; }
        ce_b[b] = a / c;
    }
}

// ---------------------------------------------------------------------------
// Kernel 4: matched-pair L1 / GIoU losses + final 4-float output.
// One block of 64 threads (one per batch).
// ---------------------------------------------------------------------------
__global__ void __launch_bounds__(64)
k_final(const float* __restrict__ pxyxy,
        const float* __restrict__ tboxes,
        const int*   __restrict__ pi,
        const int*   __restrict__ ti,
        const float* __restrict__ ce_b,
        float* __restrict__ out)
{
    __shared__ float sl1[BATCH], sgl[BATCH], sce[BATCH];
    const int b = threadIdx.x;          // 64 threads
    float l1 = 0.0f, gl = 0.0f;
    for (int n = 0; n < NT; ++n) {
        int q = pi[b * NT + n];
        int t = ti[b * NT + n];
        const float* p = pxyxy + ((size_t)(b * NQ + q)) * 4;
        const float* tb = tboxes + ((size_t)(b * NT + t)) * 4;
        float p4[4] = { p[0], p[1], p[2], p[3] };
        float t4[4] = { tb[0], tb[1], tb[2], tb[3] };
        l1 += fabsf(p4[0] - t4[0]) + fabsf(p4[1] - t4[1]) +
              fabsf(p4[2] - t4[2]) + fabsf(p4[3] - t4[3]);
        gl += 1.0f - giou_xyxy1(p4, t4);
    }
    sl1[b] = l1 * (1.0f / (NT * 4.0f));
    sgl[b] = gl * (1.0f / (float)NT);
    sce[b] = ce_b[b];
    __syncthreads();
    if (b == 0) {
        float a = 0.0f, c = 0.0f, g = 0.0f;
        for (int i = 0; i < BATCH; ++i) { a += sce[i]; c += sl1[i]; g += sgl[i]; }
        a *= (1.0f / BATCH); c *= (1.0f / BATCH); g *= (1.0f / BATCH);
        out[0] = a + 5.0f * c + 2.0f * g;
        out[1] = a;
        out[2] = c;
        out[3] = g;
    }
}

// ---------------------------------------------------------------------------
// launch
// ---------------------------------------------------------------------------
extern "C" void kernel_launch(void* const* d_in, const int* in_sizes, int n_in,
                              void* d_out, int out_size, void* d_ws, size_t ws_size,
                              hipStream_t stream)
{
    (void)in_sizes; (void)n_in; (void)out_size; (void)ws_size;
    const float* logits = (const float*)d_in[0];   // [64,900,1001]
    const float* pboxes = (const float*)d_in[1];   // [64,900,4]
    const int*   labels = (const int*)  d_in[2];   // [64,30]
    const float* tboxes = (const float*)d_in[3];   // [64,30,4]
    float* out = (float*)d_out;                    // [4]

    // workspace carve (f32/i32, all 4B aligned; total ~8.3 MB)
    float* ws_f  = (float*)d_ws;
    float* d_lse = ws_f;                           // 57600
    float* d_px  = d_lse + BATCH * NQ;             // 230400
    float* d_C   = d_px + BATCH * NQ * 4;          // 1728000
    int*   d_pi  = (int*)(d_C + (size_t)BATCH * QN);  // 1920
    int*   d_ti  = d_pi + BATCH * NT;              // 1920
    int*   d_tc  = d_ti + BATCH * NT;              // 57600
    float* d_ceb = (float*)(d_tc + BATCH * NQ);    // 64

    k_lse_cost<<<BATCH * NQ, 256, 0, stream>>>(logits, pboxes, labels, tboxes,
                                               d_lse, d_px, d_C);

    const size_t match_lds = (size_t)(QN + 32) * 4 + (32 + NT + NT + 2) * 4;
    k_match<<<BATCH, 1024, match_lds, stream>>>(d_C, labels, d_pi, d_ti, d_tc);

    k_ce<<<BATCH, 256, 0, stream>>>(logits, d_lse, d_tc, d_ceb);

    k_final<<<1, BATCH, 0, stream>>>(d_px, tboxes, d_pi, d_ti, d_ceb, out);
}